// LocalGeodesicAttention_9191230013584
// MI455X (gfx1250) — compile-verified
//
#include <hip/hip_runtime.h>
#include <math.h>

// ---------------------------------------------------------------------------
// LocalGeodesicAttention for MI455X (gfx1250, wave32).
//
// Algebraic optimization vs reference: unfold(zero_pad(x)) @ W + b ==
// gather of (x_flat @ W + b) at shifted pixels; OOB windows give exactly the
// bias vector. K/V projections shrink from [B*N*K,C]x[C,C] (59 GFLOP, 462 MB
// im2col) to [B*N,C]x[C,C] (0.6 GFLOP each). GEMMs use V_WMMA_F32_16X16X4_F32
// (full fp32 to match reference numerics; the fused problem is ~100 MB of
// L2-resident traffic at 23.3 TB/s, so fp32 matrix throughput is not the
// limiter -- VMEM issue rate is, which is why each wave amortizes the large
// operand across 8 WMMAs in registers).
// ---------------------------------------------------------------------------

typedef __attribute__((ext_vector_type(2))) float v2f;
typedef __attribute__((ext_vector_type(8))) float v8f;

#define BB      2
#define CC      128
#define HH      96
#define WW      96
#define NHEADS  4
#define WINR    7
#define MMF     16
#define HD      32
#define NN      (HH * WW)     /* 9216  */
#define KK      (WINR * WINR) /* 49    */
#define BN      (BB * NN)     /* 18432 */

static __device__ __forceinline__ v8f wmma_f32(v2f a, v2f b, v8f c) {
    return __builtin_amdgcn_wmma_f32_16x16x4_f32(false, a, false, b,
                                                 (short)0, c, false, false);
}

// ---------------------------------------------------------------------------
// Kernel 1: fused Q/K/V projection. One wave computes a 16-pixel x 128-channel
// strip of D = X_flat @ W + b. Per K-step (4), the activation fragment A is
// loaded ONCE and reused by 8 consecutive V_WMMA_F32_16X16X4_F32 against 8
// weight fragments; 8 v8f accumulators (seeded with bias) stay in registers.
//   A[m,k] = x[b, k, n]   (lane-contiguous across the 16 pixels)
//   B[k,c] = W[k*C + c]   (lane-contiguous across the 16 channels)
// ---------------------------------------------------------------------------
__global__ __launch_bounds__(128)
void qkv_wmma(const float* __restrict__ x,
              const float* __restrict__ Wq, const float* __restrict__ bq,
              const float* __restrict__ Wk, const float* __restrict__ bk,
              const float* __restrict__ Wv, const float* __restrict__ bv,
              float* __restrict__ Q, float* __restrict__ Kp, float* __restrict__ Vp)
{
    const int lane = threadIdx.x & 31;
    const int wid  = threadIdx.x >> 5;
    const int mt   = blockIdx.x * 4 + wid;   // pixel tile 0..1151
    const int pj   = blockIdx.y;             // 0=Q 1=K 2=V

    const float* W    = (pj == 0) ? Wq : (pj == 1) ? Wk : Wv;
    const float* bias = (pj == 0) ? bq : (pj == 1) ? bk : bv;
    float*       D    = (pj == 0) ? Q  : (pj == 1) ? Kp : Vp;

    const int m0  = mt * 16;                 // N=9216 % 16 == 0: no batch crossing
    const int b   = m0 / NN;
    const int n0  = m0 % NN;

    const int row = lane & 15;               // A-row (pixel) / B-col (channel)
    const int hi  = lane >> 4;               // half-wave: K {0,1} vs {2,3}
    const int ks  = hi * 2;

    v8f acc[8];
#pragma unroll
    for (int t = 0; t < 8; ++t) {
        const float bval = bias[t * 16 + row];
#pragma unroll
        for (int r = 0; r < 8; ++r) acc[t][r] = bval;
    }

    const float* xb = x + (size_t)b * CC * NN;    // x[b, k, n]
    const int n = n0 + row;

#pragma unroll 2
    for (int k0 = 0; k0 < CC; k0 += 4) {
        v2f a;                                     // A fragment: loaded once,
        a.x = xb[(size_t)(k0 + ks)     * NN + n];  // reused by 8 WMMAs
        a.y = xb[(size_t)(k0 + ks + 1) * NN + n];
        const float* w0 = W + (size_t)(k0 + ks)     * CC + row;
        const float* w1 = W + (size_t)(k0 + ks + 1) * CC + row;
#pragma unroll
        for (int t = 0; t < 8; ++t) {
            v2f bf;
            bf.x = w0[t * 16];
            bf.y = w1[t * 16];
            acc[t] = wmma_f32(a, bf, acc[t]);
        }
    }

    // D vgpr r of tile t: pixel m0 + hi*8 + r, channel t*16 + row (coalesced)
    float* Dp = D + (size_t)(m0 + hi * 8) * CC + row;
#pragma unroll
    for (int t = 0; t < 8; ++t)
#pragma unroll
        for (int r = 0; r < 8; ++r)
            Dp[(size_t)r * CC + t * 16] = acc[t][r];
}

// ---------------------------------------------------------------------------
// Kernel 2: geodesic distance precompute (head-independent, done once, not 4x).
// G[b,n,k] = sum_m (phi[b,m,n] - phi_pad[b,m,n+off_k])^2, OOB pad = 1.0.
// ---------------------------------------------------------------------------
__global__ __launch_bounds__(256)
void geo_kernel(const float* __restrict__ phi, float* __restrict__ G)
{
    const int idx = blockIdx.x * blockDim.x + threadIdx.x;
    if (idx >= BB * NN * KK) return;
    const int k  = idx % KK;
    const int bn = idx / KK;
    const int n  = bn % NN;
    const int b  = bn / NN;
    const int y  = n / WW, xq = n % WW;
    const int dy = k / WINR - WINR / 2;
    const int dx = k % WINR - WINR / 2;
    const int ny = y + dy, nx = xq + dx;
    const bool oob = ((unsigned)ny >= HH) | ((unsigned)nx >= WW);
    const int nn2 = ny * WW + nx;

    const float* pb = phi + (size_t)b * MMF * NN;
    float acc = 0.f;
#pragma unroll
    for (int m = 0; m < MMF; ++m) {
        const float c  = pb[(size_t)m * NN + n];
        const float nb = oob ? 1.0f : pb[(size_t)m * NN + nn2];
        const float d  = c - nb;
        acc += d * d;
    }
    G[idx] = acc;
}

// ---------------------------------------------------------------------------
// Kernel 3: sliding-window attention. One wave32 per (b, head, pixel);
// lane = head-dim channel (HD == 32 == wavesize). Online softmax over the 49
// taps; OOB taps use the bias vectors (exact consequence of zero-pad unfold).
// q.k dot is a 5-step wave32 butterfly reduction.
// ---------------------------------------------------------------------------
__global__ __launch_bounds__(256)
void attn_kernel(const float* __restrict__ Q, const float* __restrict__ Kp,
                 const float* __restrict__ Vp,
                 const float* __restrict__ bk, const float* __restrict__ bv,
                 const float* __restrict__ G,
                 const float* __restrict__ log_alpha,
                 const float* __restrict__ beta,
                 float* __restrict__ O)
{
    const int wave = (int)((blockIdx.x * blockDim.x + threadIdx.x) >> 5);
    const int lane = threadIdx.x & 31;
    if (wave >= BB * NHEADS * NN) return;

    const int n  = wave % NN;
    const int bh = wave / NN;
    const int h  = bh % NHEADS;
    const int b  = bh / NHEADS;
    const int y  = n / WW, xx = n % WW;

    const float scale      = 0.17677669529663687f;  // HD^-0.5
    const float normalizer = 5.656854249492381f;    // sqrt(2*M)
    const float coef = -__expf(log_alpha[0]) * beta[h] / normalizer;

    const size_t base = (size_t)(b * NN + n) * CC + h * HD;
    const float  q    = Q[base + lane] * scale;
    const float  bkv  = bk[h * HD + lane];
    const float  bvv  = bv[h * HD + lane];

    const float* Gp = G  + (size_t)(b * NN + n) * KK;
    const float* Kb = Kp + (size_t)b * NN * CC + h * HD + lane;
    const float* Vb = Vp + (size_t)b * NN * CC + h * HD + lane;

    float mmax  = -3.0e38f;
    float denom = 0.f;
    float o     = 0.f;

    for (int k = 0; k < KK; ++k) {
        const int dy = k / WINR - 3;
        const int dx = k % WINR - 3;
        const int ny = y + dy, nx = xx + dx;
        const bool oob = ((unsigned)ny >= HH) | ((unsigned)nx >= WW);
        const int nn2 = ny * WW + nx;

        const float kd = oob ? bkv : Kb[(size_t)nn2 * CC];
        const float vd = oob ? bvv : Vb[(size_t)nn2 * CC];

        float p = q * kd;
#pragma unroll
        for (int off = 16; off > 0; off >>= 1)
            p += __shfl_xor(p, off, 32);           // wave32 butterfly reduce

        const float logit = p + coef * Gp[k];
        const float mnew  = fmaxf(mmax, logit);
        const float corr  = __expf(mmax - mnew);
        const float e     = __expf(logit - mnew);
        denom = denom * corr + e;
        o     = o * corr + e * vd;
        mmax  = mnew;
    }
    O[base + lane] = o / denom;
}

// ---------------------------------------------------------------------------
// Kernel 4: output projection, computed transposed (D = Wo^T x O^T) so the
// [B,C,H,W] store is lane-contiguous. One wave computes 128 channels x 16
// pixels: the B fragment (O pixel data, the big operand) is one float2 load
// per K-step, reused by 8 WMMAs against 8 Wo^T fragments.
// ---------------------------------------------------------------------------
__global__ __launch_bounds__(128)
void oproj_wmma(const float* __restrict__ O, const float* __restrict__ Wo,
                const float* __restrict__ bo, float* __restrict__ out)
{
    const int lane = threadIdx.x & 31;
    const int wid  = threadIdx.x >> 5;
    const int pt   = blockIdx.x * 4 + wid;   // pixel tile 0..1151

    const int p0 = pt * 16;
    const int b  = p0 / NN;
    const int n0 = p0 % NN;

    const int row = lane & 15;
    const int hi  = lane >> 4;
    const int ks  = hi * 2;

    v8f acc[8];
#pragma unroll
    for (int t = 0; t < 8; ++t)
#pragma unroll
        for (int r = 0; r < 8; ++r)
            acc[t][r] = bo[t * 16 + hi * 8 + r];

    const float* Brow = O + (size_t)(p0 + row) * CC;   // this lane's pixel row

#pragma unroll 2
    for (int k0 = 0; k0 < CC; k0 += 4) {
        const float2 tv = *(const float2*)(Brow + k0 + ks);  // B fragment: one
        v2f bf; bf.x = tv.x; bf.y = tv.y;                    // 8B load, 8 WMMAs
        const float* w0 = Wo + (size_t)(k0 + ks)     * CC + row;
        const float* w1 = Wo + (size_t)(k0 + ks + 1) * CC + row;
#pragma unroll
        for (int t = 0; t < 8; ++t) {
            v2f a;                     // A[c,k] = Wo[k, t*16 + c]
            a.x = w0[t * 16];
            a.y = w1[t * 16];
            acc[t] = wmma_f32(a, bf, acc[t]);
        }
    }

    // D vgpr r of tile t: channel t*16 + hi*8 + r, pixel p0 + row (coalesced)
    float* op = out + (size_t)b * CC * NN + (size_t)(hi * 8) * NN + n0 + row;
#pragma unroll
    for (int t = 0; t < 8; ++t)
#pragma unroll
        for (int r = 0; r < 8; ++r)
            op[(size_t)(t * 16 + r) * NN] = acc[t][r];
}

// ---------------------------------------------------------------------------
extern "C" void kernel_launch(void* const* d_in, const int* in_sizes, int n_in,
                              void* d_out, int out_size, void* d_ws, size_t ws_size,
                              hipStream_t stream)
{
    (void)in_sizes; (void)n_in; (void)out_size; (void)ws_size;

    const float* x    = (const float*)d_in[0];
    const float* phi  = (const float*)d_in[1];
    const float* Wq   = (const float*)d_in[2];
    const float* bq   = (const float*)d_in[3];
    const float* Wk   = (const float*)d_in[4];
    const float* bk   = (const float*)d_in[5];
    const float* Wv   = (const float*)d_in[6];
    const float* bv   = (const float*)d_in[7];
    const float* Wo   = (const float*)d_in[8];
    const float* bo   = (const float*)d_in[9];
    const float* la   = (const float*)d_in[10];
    const float* beta = (const float*)d_in[11];
    float* out = (float*)d_out;

    // Workspace: Q,K,V,O each BN*CC floats (9.4 MB), G = BN*KK floats (3.6 MB)
    float* Q  = (float*)d_ws;
    float* Kp = Q  + (size_t)BN * CC;
    float* Vp = Kp + (size_t)BN * CC;
    float* Oo = Vp + (size_t)BN * CC;
    float* G  = Oo + (size_t)BN * CC;

    qkv_wmma<<<dim3(BN / 16 / 4, 3), 128, 0, stream>>>(x, Wq, bq, Wk, bk,
                                                       Wv, bv, Q, Kp, Vp);

    geo_kernel<<<(BB * NN * KK + 255) / 256, 256, 0, stream>>>(phi, G);

    attn_kernel<<<(BB * NHEADS * NN) / 8, 256, 0, stream>>>(Q, Kp, Vp, bk, bv,
                                                            G, la, beta, Oo);

    oproj_wmma<<<BN / 16 / 4, 128, 0, stream>>>(Oo, Wo, bo, out);
}